// DimEncoder_19894288515585
// MI455X (gfx1250) — compile-verified
//
#include <hip/hip_runtime.h>
#include <math.h>

// ---------------------------------------------------------------------------
// Types for CDNA5 WMMA (wave32). f32 16x16x4: A/B fragments are 2 VGPRs/lane,
// C/D accumulator is 8 VGPRs/lane.
// ---------------------------------------------------------------------------
typedef __attribute__((ext_vector_type(2))) float v2f;
typedef __attribute__((ext_vector_type(8))) float v8f;

#define LEAKY_SLOPE 0.01f
#define SCALE_PARAM 3.0f

__device__ __forceinline__ float leaky(float x) {
    return x >= 0.0f ? x : LEAKY_SLOPE * x;
}

// ---------------------------------------------------------------------------
// WMMA f32 GEMM:  C[M x N] = act(A)[M x K] @ B[K x N] (+ bias[N])
// BM=64, BN=128, BK=32; 256 threads = 8 waves in a 2(M) x 4(N) grid.
// Each wave computes a 32x32 tile = 4 accumulators (2 A-frags x 2 B-frags,
// 4 v_wmma per k-step -> 1:1 LDS-fragment:WMMA ratio).
// Requires: K % 32 == 0, N % 128 == 0 (true here: K in {1024,256}, N in {256,128}).
// ---------------------------------------------------------------------------
template <bool LEAKY_A, bool HAS_BIAS>
__global__ __launch_bounds__(256) void gemm_wmma_f32(
    const float* __restrict__ A, const float* __restrict__ B,
    const float* __restrict__ bias, float* __restrict__ C,
    int M, int N, int K) {
  constexpr int BM = 64, BN = 128, BK = 32;
  __shared__ float As[BK][BM + 1];  // transposed: As[k][m]
  __shared__ float Bs[BK][BN + 1];  // row-major:  Bs[k][n]

  const int tid  = threadIdx.x;
  const int wave = tid >> 5;
  const int lane = tid & 31;
  const int half = lane >> 4;     // 0: lanes 0-15, 1: lanes 16-31
  const int ml   = lane & 15;
  const int wm   = (wave & 1) * 32;   // wave's row offset inside block tile
  const int wn   = (wave >> 1) * 32;  // wave's col offset inside block tile
  const int m0   = blockIdx.x * BM;
  const int n0   = blockIdx.y * BN;

  v8f acc00 = {}, acc01 = {}, acc10 = {}, acc11 = {};

  for (int kt = 0; kt < K; kt += BK) {
    // ---- load A tile (64 x 32) transposed into LDS, float4 global reads ----
    {
      const int r  = tid >> 3;         // 0..31, two passes cover 64 rows
      const int c4 = (tid & 7) * 4;    // 0..28
#pragma unroll
      for (int rr = 0; rr < 2; ++rr) {
        const int row = m0 + r + rr * 32;
        float4 val = make_float4(0.f, 0.f, 0.f, 0.f);
        if (row < M)
          val = *(const float4*)&A[(long long)row * K + kt + c4];
        if (LEAKY_A) {
          val.x = leaky(val.x); val.y = leaky(val.y);
          val.z = leaky(val.z); val.w = leaky(val.w);
        }
        As[c4 + 0][r + rr * 32] = val.x;
        As[c4 + 1][r + rr * 32] = val.y;
        As[c4 + 2][r + rr * 32] = val.z;
        As[c4 + 3][r + rr * 32] = val.w;
      }
    }
    // ---- load B tile (32 x 128), K%32==0 and N%128==0 so no guards ----
    {
      const int r  = tid >> 5;         // 0..7, four passes cover 32 rows
      const int c4 = (tid & 31) * 4;   // 0..124
#pragma unroll
      for (int rr = 0; rr < 4; ++rr) {
        const int kr = kt + r + rr * 8;
        float4 val = *(const float4*)&B[(long long)kr * N + n0 + c4];
        Bs[r + rr * 8][c4 + 0] = val.x;
        Bs[r + rr * 8][c4 + 1] = val.y;
        Bs[r + rr * 8][c4 + 2] = val.z;
        Bs[r + rr * 8][c4 + 3] = val.w;
      }
    }
    __syncthreads();

    // ---- WMMA over the K-tile: 8 k-steps of 4, 2x2 subtiles per wave ----
#pragma unroll
    for (int kk = 0; kk < BK; kk += 4) {
      const int ka = kk + 2 * half;  // lanes 0-15: K=kk,kk+1; lanes 16-31: kk+2,kk+3
      v2f a0, a1, b0, b1;
      a0.x = As[ka][wm + ml];
      a0.y = As[ka + 1][wm + ml];
      a1.x = As[ka][wm + 16 + ml];
      a1.y = As[ka + 1][wm + 16 + ml];
      b0.x = Bs[ka][wn + ml];
      b0.y = Bs[ka + 1][wn + ml];
      b1.x = Bs[ka][wn + 16 + ml];
      b1.y = Bs[ka + 1][wn + 16 + ml];
      acc00 = __builtin_amdgcn_wmma_f32_16x16x4_f32(
          false, a0, false, b0, (short)0, acc00, false, false);
      acc01 = __builtin_amdgcn_wmma_f32_16x16x4_f32(
          false, a0, false, b1, (short)0, acc01, false, false);
      acc10 = __builtin_amdgcn_wmma_f32_16x16x4_f32(
          false, a1, false, b0, (short)0, acc10, false, false);
      acc11 = __builtin_amdgcn_wmma_f32_16x16x4_f32(
          false, a1, false, b1, (short)0, acc11, false, false);
    }
    __syncthreads();
  }

  // ---- store D: VGPR r -> row (8*half + r), col = ml within each 16x16 tile ----
#pragma unroll
  for (int mi = 0; mi < 2; ++mi) {
#pragma unroll
    for (int r = 0; r < 8; ++r) {
      const int row = m0 + wm + mi * 16 + half * 8 + r;
      if (row < M) {
        const int c0 = n0 + wn + ml;
        const int c1 = c0 + 16;
        float o0 = (mi == 0) ? acc00[r] : acc10[r];
        float o1 = (mi == 0) ? acc01[r] : acc11[r];
        if (HAS_BIAS) { o0 += bias[c0]; o1 += bias[c1]; }
        C[(long long)row * N + c0] = o0;
        C[(long long)row * N + c1] = o1;
      }
    }
  }
}

// ---------------------------------------------------------------------------
// GCN normalization: deg (with self loop) -> dinv = rsqrt(deg)
// ---------------------------------------------------------------------------
__global__ void k_deg_init(float* deg, int n) {
  int i = blockIdx.x * blockDim.x + threadIdx.x;
  if (i < n) deg[i] = 1.0f;  // self loop
}

__global__ void k_deg_count(const long long* __restrict__ ei, float* deg, int E) {
  int e = blockIdx.x * blockDim.x + threadIdx.x;
  if (e < E) atomicAdd(&deg[(int)ei[(long long)E + e]], 1.0f);  // dst
}

__global__ void k_dinv(float* deg, int n) {
  int i = blockIdx.x * blockDim.x + threadIdx.x;
  if (i < n) {
    float d = deg[i];
    deg[i] = d > 0.0f ? rsqrtf(d) : 0.0f;
  }
}

// h[i][c] = b[c] + hp[i][c]*dinv[i]^2   (self-loop term + bias)
__global__ void k_h_init(const float* __restrict__ hp,
                         const float* __restrict__ dinv,
                         const float* __restrict__ b, float* __restrict__ h,
                         int total) {
  int idx = blockIdx.x * blockDim.x + threadIdx.x;
  if (idx < total) {
    int i = idx >> 8;       // H_MID = 256
    int c = idx & 255;
    float di = dinv[i];
    h[idx] = b[c] + hp[idx] * di * di;
  }
}

// wave-per-edge: h[dst] += hp[src] * dinv[src]*dinv[dst]
__global__ __launch_bounds__(256) void k_h_scatter(
    const long long* __restrict__ ei, const float* __restrict__ hp,
    const float* __restrict__ dinv, float* __restrict__ h, int E) {
  int e = (blockIdx.x * blockDim.x + threadIdx.x) >> 5;
  int lane = threadIdx.x & 31;
  if (e >= E) return;
  int s = (int)ei[e];
  int d = (int)ei[(long long)E + e];
  float w = dinv[s] * dinv[d];
  const float* hs = hp + (long long)s * 256;
  float* hd = h + (long long)d * 256;
#pragma unroll
  for (int c = lane; c < 256; c += 32) atomicAdd(&hd[c], hs[c] * w);
}

// ---------------------------------------------------------------------------
// alpha[e] = dot(q[dst], k[src]) / sqrt(128); deterministic per-block partials
// ---------------------------------------------------------------------------
__global__ __launch_bounds__(256) void k_alpha(
    const long long* __restrict__ ei, const float* __restrict__ q,
    const float* __restrict__ kk, float* __restrict__ alpha,
    float* __restrict__ partial, int E) {
  __shared__ float sa[8];
  int wid = threadIdx.x >> 5, lane = threadIdx.x & 31;
  int e = blockIdx.x * 8 + wid;
  float a = 0.0f;
  if (e < E) {
    int s = (int)ei[e];
    int d = (int)ei[(long long)E + e];
    const float* qd = q + (long long)d * 128;
    const float* ks = kk + (long long)s * 128;
    float acc = 0.0f;
#pragma unroll
    for (int c = lane; c < 128; c += 32) acc += qd[c] * ks[c];
#pragma unroll
    for (int o = 16; o; o >>= 1) acc += __shfl_xor(acc, o, 32);
    a = acc * 0.08838834764831845f;  // 1/sqrt(128)
    if (lane == 0) alpha[e] = a;
  }
  if (lane == 0) sa[wid] = (e < E) ? a : 0.0f;
  __syncthreads();
  if (threadIdx.x == 0) {
    float s1 = 0.f, s2 = 0.f;
#pragma unroll
    for (int i = 0; i < 8; ++i) { float v = sa[i]; s1 += v; s2 += v * v; }
    partial[2 * blockIdx.x]     = s1;
    partial[2 * blockIdx.x + 1] = s2;
  }
}

// single block; fixed-order strided accumulation -> deterministic mean/std
__global__ __launch_bounds__(256) void k_stats(const float* __restrict__ partial,
                                               int nblk, int E,
                                               float* __restrict__ stats) {
  __shared__ float s1s[256], s2s[256];
  int t = threadIdx.x;
  float s1 = 0.f, s2 = 0.f;
  for (int i = t; i < nblk; i += 256) { s1 += partial[2 * i]; s2 += partial[2 * i + 1]; }
  s1s[t] = s1; s2s[t] = s2;
  __syncthreads();
  for (int o = 128; o; o >>= 1) {
    if (t < o) { s1s[t] += s1s[t + o]; s2s[t] += s2s[t + o]; }
    __syncthreads();
  }
  if (t == 0) {
    float mean = s1s[0] / (float)E;
    float var  = (s2s[0] - (float)E * mean * mean) / (float)(E - 1);  // unbiased
    float sd   = sqrtf(fmaxf(var, 0.0f));
    stats[0] = mean;
    stats[1] = SCALE_PARAM / sd;
  }
}

// wave-per-edge: out[dst] += v[src] * sigmoid((alpha-mean)*scale/std)
__global__ __launch_bounds__(256) void k_message(
    const long long* __restrict__ ei, const float* __restrict__ v,
    const float* __restrict__ alpha, const float* __restrict__ stats,
    float* __restrict__ out, int E) {
  int e = (blockIdx.x * blockDim.x + threadIdx.x) >> 5;
  int lane = threadIdx.x & 31;
  if (e >= E) return;
  int s = (int)ei[e];
  int d = (int)ei[(long long)E + e];
  float z = (alpha[e] - stats[0]) * stats[1];
  float coef = 1.0f / (1.0f + expf(-z));
  const float* vs = v + (long long)s * 128;
  float* od = out + (long long)d * 128;
#pragma unroll
  for (int c = lane; c < 128; c += 32) atomicAdd(&od[c], vs[c] * coef);
}

// ---------------------------------------------------------------------------
// Orchestration
// ---------------------------------------------------------------------------
extern "C" void kernel_launch(void* const* d_in, const int* in_sizes, int n_in,
                              void* d_out, int out_size, void* d_ws, size_t ws_size,
                              hipStream_t stream) {
  const float*      x  = (const float*)d_in[0];
  const long long*  ei = (const long long*)d_in[1];  // int64 [2, E]
  const float* Wg = (const float*)d_in[2];
  const float* bg = (const float*)d_in[3];
  const float* Wq = (const float*)d_in[4];
  const float* bq = (const float*)d_in[5];
  const float* Wk = (const float*)d_in[6];
  const float* bk = (const float*)d_in[7];
  const float* Wv = (const float*)d_in[8];
  const float* bv = (const float*)d_in[9];
  const float* Ws = (const float*)d_in[10];
  const float* bs = (const float*)d_in[11];
  float* out = (float*)d_out;

  const int H = in_sizes[3];              // 256
  const int D = in_sizes[5];              // 128
  const int F = in_sizes[2] / H;          // 1024
  const int N = in_sizes[0] / F;          // 20000
  const int E = in_sizes[1] / 2;          // 320000

  // workspace carve-up (floats)
  float* w      = (float*)d_ws;
  float* hp     = w; w += (size_t)N * H;      // pre-aggregation x@Wg
  float* h      = w; w += (size_t)N * H;      // aggregated (pre-LeakyReLU)
  float* dinv   = w; w += (size_t)N;          // deg -> rsqrt(deg)
  float* qb     = w; w += (size_t)N * D;
  float* kb     = w; w += (size_t)N * D;
  float* vb     = w; w += (size_t)N * D;
  float* alpha  = w; w += (size_t)E;
  const int nablk = (E + 7) / 8;
  float* partial = w; w += (size_t)2 * nablk;
  float* stats   = w; w += 2;

  const int TB = 256;
  // --- GCN normalization ---
  k_deg_init<<<(N + TB - 1) / TB, TB, 0, stream>>>(dinv, N);
  k_deg_count<<<(E + TB - 1) / TB, TB, 0, stream>>>(ei, dinv, E);
  k_dinv<<<(N + TB - 1) / TB, TB, 0, stream>>>(dinv, N);

  // --- hp = x @ W_gcn (WMMA f32) ---
  {
    dim3 grid((N + 63) / 64, H / 128);
    gemm_wmma_f32<false, false><<<grid, TB, 0, stream>>>(x, Wg, nullptr, hp, N, H, F);
  }

  // --- h = b + hp*dinv^2 + scatter(hp[src]*norm) ---
  k_h_init<<<((size_t)N * H + TB - 1) / TB, TB, 0, stream>>>(hp, dinv, bg, h, N * H);
  k_h_scatter<<<((size_t)E * 32 + TB - 1) / TB, TB, 0, stream>>>(ei, hp, dinv, h, E);

  // --- q/k/v/skip GEMMs (LeakyReLU fused into A-tile load) ---
  {
    dim3 grid((N + 63) / 64, D / 128);
    gemm_wmma_f32<true, true><<<grid, TB, 0, stream>>>(h, Wq, bq, qb, N, D, H);
    gemm_wmma_f32<true, true><<<grid, TB, 0, stream>>>(h, Wk, bk, kb, N, D, H);
    gemm_wmma_f32<true, true><<<grid, TB, 0, stream>>>(h, Wv, bv, vb, N, D, H);
    gemm_wmma_f32<true, true><<<grid, TB, 0, stream>>>(h, Ws, bs, out, N, D, H);  // skip term -> out
  }

  // --- attention scores, normalization stats, message scatter ---
  k_alpha<<<nablk, TB, 0, stream>>>(ei, qb, kb, alpha, partial, E);
  k_stats<<<1, TB, 0, stream>>>(partial, nablk, E, stats);
  k_message<<<((size_t)E * 32 + TB - 1) / TB, TB, 0, stream>>>(ei, vb, alpha, stats, out, E);
}